// MultiViewGaussianSplatTrainer_69913477644664
// MI455X (gfx1250) — compile-verified
//
#include <hip/hip_runtime.h>
#include <math.h>

typedef __attribute__((ext_vector_type(2))) float v2f;
typedef __attribute__((ext_vector_type(8))) float v8f;
typedef unsigned int u32x4 __attribute__((ext_vector_type(4)));
typedef int          i32x4 __attribute__((ext_vector_type(4)));
typedef int          i32x8 __attribute__((ext_vector_type(8)));

#define SORT_N 2048
#define IMG_H  128
#define IMG_W  128
#define NEAR_P 0.1f

// ---------------------------------------------------------------------------
// Phase 1: per-gaussian preprocessing.
// Writes unsorted SoA arrays into ws:
//   [0]u [1]v [2]cA [3]cB [4]cC [5]op [6]r [7]g [8]b [9]zkey   (each length n)
// ---------------------------------------------------------------------------
__global__ __launch_bounds__(256) void gs_preprocess(
    const float* __restrict__ means, const float* __restrict__ logs,
    const float* __restrict__ rots,  const float* __restrict__ cols,
    const float* __restrict__ opac,  const float* __restrict__ intr,
    const float* __restrict__ c2w,   float* __restrict__ ws, int n)
{
  int i = blockIdx.x * blockDim.x + threadIdx.x;
  if (i >= n) return;

  // quaternion -> rotation matrix
  float qw = rots[i*4+0], qx = rots[i*4+1], qy = rots[i*4+2], qz = rots[i*4+3];
  float qn = fmaxf(sqrtf(qw*qw + qx*qx + qy*qy + qz*qz), 1e-8f);
  qw /= qn; qx /= qn; qy /= qn; qz /= qn;
  float xx = qx*qx, yy = qy*qy, zz = qz*qz;
  float xy = qx*qy, xz = qx*qz, yz = qy*qz;
  float wx = qw*qx, wy = qw*qy, wz = qw*qz;
  float R[3][3] = {
    {1.0f-2.0f*(yy+zz), 2.0f*(xy-wz),      2.0f*(xz+wy)},
    {2.0f*(xy+wz),      1.0f-2.0f*(xx+zz), 2.0f*(yz-wx)},
    {2.0f*(xz-wy),      2.0f*(yz+wx),      1.0f-2.0f*(xx+yy)}};

  float s2[3];
#pragma unroll
  for (int k = 0; k < 3; ++k) s2[k] = __expf(2.0f * logs[i*3+k]);

  float cov3[3][3];
#pragma unroll
  for (int a = 0; a < 3; ++a)
#pragma unroll
    for (int b = 0; b < 3; ++b) {
      float s = 0.0f;
#pragma unroll
      for (int k = 0; k < 3; ++k) s += R[a][k] * s2[k] * R[b][k];
      cov3[a][b] = s + ((a == b) ? 1e-6f : 0.0f);
    }

  // world -> camera
  float Rwc[3][3], tcw[3], twc[3];
#pragma unroll
  for (int a = 0; a < 3; ++a)
#pragma unroll
    for (int b = 0; b < 3; ++b) Rwc[a][b] = c2w[b*4 + a];
#pragma unroll
  for (int a = 0; a < 3; ++a) tcw[a] = c2w[a*4 + 3];
#pragma unroll
  for (int a = 0; a < 3; ++a) {
    float s = 0.0f;
#pragma unroll
    for (int b = 0; b < 3; ++b) s += Rwc[a][b] * tcw[b];
    twc[a] = -s;
  }
  float mn[3] = {means[i*3+0], means[i*3+1], means[i*3+2]};
  float p[3];
#pragma unroll
  for (int a = 0; a < 3; ++a) {
    float s = twc[a];
#pragma unroll
    for (int b = 0; b < 3; ++b) s += Rwc[a][b] * mn[b];
    p[a] = s;
  }

  float z     = p[2];
  float valid = (z > NEAR_P) ? 1.0f : 0.0f;
  float zc    = fmaxf(z, NEAR_P);
  float fx = intr[0], fy = intr[4], cx = intr[2], cy = intr[5];
  float u = fx * p[0] / zc + cx;
  float v = fy * p[1] / zc + cy;

  float J[2][3] = {
    {fx/zc, 0.0f,  -fx*p[0]/(zc*zc)},
    {0.0f,  fy/zc, -fy*p[1]/(zc*zc)}};

  // cov_cam = Rwc * cov3 * Rwc^T
  float tmp[3][3], covc[3][3];
#pragma unroll
  for (int a = 0; a < 3; ++a)
#pragma unroll
    for (int b = 0; b < 3; ++b) {
      float s = 0.0f;
#pragma unroll
      for (int k = 0; k < 3; ++k) s += Rwc[a][k] * cov3[k][b];
      tmp[a][b] = s;
    }
#pragma unroll
  for (int a = 0; a < 3; ++a)
#pragma unroll
    for (int b = 0; b < 3; ++b) {
      float s = 0.0f;
#pragma unroll
      for (int k = 0; k < 3; ++k) s += tmp[a][k] * Rwc[b][k];
      covc[a][b] = s;
    }

  // cov2d = J * cov_cam * J^T + 0.3 I
  float M[2][3];
#pragma unroll
  for (int a = 0; a < 2; ++a)
#pragma unroll
    for (int b = 0; b < 3; ++b) {
      float s = 0.0f;
#pragma unroll
      for (int k = 0; k < 3; ++k) s += J[a][k] * covc[k][b];
      M[a][b] = s;
    }
  float c00 = 0.3f, c01 = 0.0f, c11 = 0.3f;
#pragma unroll
  for (int k = 0; k < 3; ++k) {
    c00 += M[0][k] * J[0][k];
    c01 += M[0][k] * J[1][k];
    c11 += M[1][k] * J[1][k];
  }
  float det = fmaxf(c00*c11 - c01*c01, 1e-12f);
  float cA = c11/det, cB = -c01/det, cC = c00/det;

  float op = valid / (1.0f + __expf(-opac[i]));

  ws[0*n+i] = u;  ws[1*n+i] = v;
  ws[2*n+i] = cA; ws[3*n+i] = cB; ws[4*n+i] = cC;
  ws[5*n+i] = op;
  ws[6*n+i] = fminf(fmaxf(cols[i*3+0], 0.0f), 1.0f);
  ws[7*n+i] = fminf(fmaxf(cols[i*3+1], 0.0f), 1.0f);
  ws[8*n+i] = fminf(fmaxf(cols[i*3+2], 0.0f), 1.0f);
  ws[9*n+i] = (valid > 0.0f) ? zc : __builtin_inff();
}

// ---------------------------------------------------------------------------
// Phase 2: single-workgroup bitonic depth sort (2048 keys, 16 KB LDS) and
// gather of WMMA A-matrix coefficients in sorted order. The sorted table is
// one contiguous 10*n-float block at ws+10n:
//   +0n..6n : G0..G5 (quadratic-form coeffs for [x^2, y^2, xy, x, y, 1])
//   +6n : op   +7n : r   +8n : g   +9n : b
// ---------------------------------------------------------------------------
__global__ __launch_bounds__(1024) void gs_sort_gather(float* __restrict__ ws, int n)
{
  __shared__ float key[SORT_N];
  __shared__ int   idx[SORT_N];
  int tid = threadIdx.x;

  for (int i = tid; i < SORT_N; i += 1024) {
    key[i] = (i < n) ? ws[9*n + i] : __builtin_inff();
    idx[i] = i;
  }
  __syncthreads();

  for (int size = 2; size <= SORT_N; size <<= 1) {
    for (int stride = size >> 1; stride > 0; stride >>= 1) {
      for (int i = tid; i < SORT_N; i += 1024) {
        int j = i ^ stride;
        if (j > i) {
          bool up = ((i & size) == 0);
          float ki = key[i], kj = key[j];
          bool sw = up ? (ki > kj) : (ki < kj);
          if (sw) {
            key[i] = kj; key[j] = ki;
            int t = idx[i]; idx[i] = idx[j]; idx[j] = t;
          }
        }
      }
      __syncthreads();
    }
  }

  float* G = ws + 10*(size_t)n;
  for (int i = tid; i < n; i += 1024) {
    int s = idx[i];
    float u  = ws[0*n+s], v  = ws[1*n+s];
    float cA = ws[2*n+s], cB = ws[3*n+s], cC = ws[4*n+s];
    G[0*n+i] = -0.5f * cA;                                    // * x^2
    G[1*n+i] = -0.5f * cC;                                    // * y^2
    G[2*n+i] = -cB;                                           // * x*y
    G[3*n+i] = cA*u + cB*v;                                   // * x
    G[4*n+i] = cC*v + cB*u;                                   // * y
    G[5*n+i] = -0.5f*(cA*u*u + cC*v*v) - cB*u*v;              // * 1
    G[6*n+i] = ws[5*n+s];                                     // op
    G[7*n+i] = ws[6*n+s];                                     // r
    G[8*n+i] = ws[7*n+s];                                     // g
    G[9*n+i] = ws[8*n+s];                                     // b
  }
}

// ---------------------------------------------------------------------------
// TDM: one descriptor copies the whole sorted table (10*n f32, 1-D tile)
// from global to LDS offset 0.  D# fields per CDNA5 ISA ch.8:
//   group0: count=1 | lds_addr | global_addr[56:0] | type=2
//   group1: data_size=2 (4B), tensor_dim0=tile_dim0=elems, dim1=1,
//           tensor_dim0_stride=elems
//   groups 2/3: zero (<=2-D tensor)
// ---------------------------------------------------------------------------
static __device__ __forceinline__ void tdm_load_1d(const void* src,
                                                   unsigned lds_byte_off,
                                                   int elems)
{
  unsigned long long ga = (unsigned long long)(uintptr_t)src;
  u32x4 g0;
  g0[0] = 1u;                                             // count=1 (user D#)
  g0[1] = lds_byte_off;                                   // lds_addr
  g0[2] = (unsigned)(ga & 0xFFFFFFFFull);                 // global_addr[31:0]
  g0[3] = (unsigned)((ga >> 32) & 0x01FFFFFFull)          // global_addr[56:32]
        | 0x80000000u;                                    // type=2 ("image")
  i32x8 g1;
  g1[0] = 0x20000;                                        // data_size=2 -> 4B
  g1[1] = (elems & 0xFFFF) << 16;                         // tensor_dim0[15:0]
  g1[2] = (int)(((unsigned)elems >> 16) & 0xFFFFu)        // tensor_dim0[31:16]
        | (1 << 16);                                      // tensor_dim1 = 1
  g1[3] = (elems & 0xFFFF) << 16;                         // tile_dim0 (<=65535)
  g1[4] = 1;                                              // tile_dim1=1, tile_dim2=0
  g1[5] = elems;                                          // tensor_dim0_stride[31:0]
  g1[6] = 0;                                              // stride hi / dim1_stride lo
  g1[7] = 0;
  i32x4 z4 = {0, 0, 0, 0};
#if defined(__clang_major__) && (__clang_major__ >= 23)
  i32x8 z8 = {0, 0, 0, 0, 0, 0, 0, 0};
  __builtin_amdgcn_tensor_load_to_lds(g0, g1, z4, z4, z8, 0);
#else
  __builtin_amdgcn_tensor_load_to_lds(g0, g1, z4, z4, 0);
#endif
}

// ---------------------------------------------------------------------------
// Phase 3: render. One wave = one 16-pixel strip; chunks of 16 gaussians.
// The 80 KB sorted table is staged once per workgroup into LDS by the TDM,
// so the hot loop reads only LDS (half-wave-uniform -> bank broadcast).
// power[16g x 16p] via two V_WMMA_F32_16X16X4_F32 (rank-6 quadratic form),
// then per-half-wave sequential alpha compositing + shfl_xor(16) merge.
// ---------------------------------------------------------------------------
static __device__ __forceinline__ v8f wmma4(v2f a, v2f b, v8f c) {
  return __builtin_amdgcn_wmma_f32_16x16x4_f32(
      /*neg_a=*/false, a, /*neg_b=*/false, b,
      /*c_mod=*/(short)0, c, /*reuse_a=*/false, /*reuse_b=*/false);
}

__global__ __launch_bounds__(256) void gs_render(
    const float* __restrict__ Gall, float* __restrict__ out, int n)
{
  extern __shared__ float smem[];   // 10*n floats: G0..G5, op, r, g, b

  if (threadIdx.x < 32) {
    tdm_load_1d(Gall, 0u, 10 * n);
    __builtin_amdgcn_s_wait_tensorcnt((short)0);
  }
  __syncthreads();

  int lane = threadIdx.x & 31;
  int wave = threadIdx.x >> 5;
  int gw   = blockIdx.x * (blockDim.x >> 5) + wave;
  int row  = gw >> 3;            // 0..127
  int ct   = gw & 7;             // 16-pixel column tile
  int m    = lane & 15;
  int half = lane >> 4;          // 0: K=0,1 / gaussians 0-7; 1: K=2,3 / gaussians 8-15
  int px   = ct*16 + m;

  float x = (float)px, y = (float)row;
  float f0 = x*x, f1 = y*y, f2 = x*y, f3 = x, f4 = y, f5 = 1.0f;

  // B matrix (pixel features): VGPR0 = {K0|K2}, VGPR1 = {K1|K3}
  v2f b0, b1;
  b0.x = half ? f2 : f0;
  b0.y = half ? f3 : f1;
  b1.x = half ? 0.0f : f4;
  b1.y = half ? 0.0f : f5;

  int oA0 = half ? 2*n : 0*n;
  int oA1 = half ? 3*n : 1*n;

  float Tpix = 1.0f, accr = 0.0f, accg = 0.0f, accb = 0.0f;

  for (int base = 0; base < n; base += 16) {
    int g = base + m;
    v2f a0, a1;
    a0.x = smem[oA0 + g];
    a0.y = smem[oA1 + g];
    float c4 = smem[4*n + g], c5 = smem[5*n + g];
    a1.x = half ? 0.0f : c4;
    a1.y = half ? 0.0f : c5;

    v8f d = {0.f,0.f,0.f,0.f,0.f,0.f,0.f,0.f};
    d = wmma4(a0, b0, d);   // K = 0..3  : x^2, y^2, xy, x
    d = wmma4(a1, b1, d);   // K = 4..5  : y, 1 (K=6,7 zero-padded)

    int gb = base + (half << 3);
    float Tloc = 1.0f, cr = 0.0f, cg = 0.0f, cb = 0.0f;
#pragma unroll
    for (int k = 0; k < 8; ++k) {
      float pw    = fminf(d[k], 0.0f);
      float alpha = fminf(smem[6*n + gb + k] * __expf(pw), 0.99f);
      float w     = alpha * Tloc;
      Tloc       *= (1.0f - alpha);
      cr += w * smem[7*n + gb + k];
      cg += w * smem[8*n + gb + k];
      cb += w * smem[9*n + gb + k];
    }

    // merge the two depth halves (lanes L and L+16 hold the same pixel)
    float Th  = __shfl_xor(Tloc, 16);
    float crh = __shfl_xor(cr,  16);
    float cgh = __shfl_xor(cg,  16);
    float cbh = __shfl_xor(cb,  16);
    accr += Tpix * (cr + Tloc * crh);
    accg += Tpix * (cg + Tloc * cgh);
    accb += Tpix * (cb + Tloc * cbh);
    Tpix *= Tloc * Th;
  }

  if (half == 0) {
    int o = (row * IMG_W + px) * 3;
    out[o+0] = accr;
    out[o+1] = accg;
    out[o+2] = accb;
  }
}

// ---------------------------------------------------------------------------
extern "C" void kernel_launch(void* const* d_in, const int* in_sizes, int n_in,
                              void* d_out, int out_size, void* d_ws, size_t ws_size,
                              hipStream_t stream)
{
  const float* means = (const float*)d_in[0];
  const float* logs  = (const float*)d_in[1];
  const float* rots  = (const float*)d_in[2];
  const float* cols  = (const float*)d_in[3];
  const float* opac  = (const float*)d_in[4];
  const float* intr  = (const float*)d_in[5];
  const float* c2w   = (const float*)d_in[6];
  int n = in_sizes[4];                 // number of gaussians (2048)
  float* ws = (float*)d_ws;            // uses 20*n floats (~160 KB)

  gs_preprocess<<<(n + 255) / 256, 256, 0, stream>>>(
      means, logs, rots, cols, opac, intr, c2w, ws, n);

  gs_sort_gather<<<1, 1024, 0, stream>>>(ws, n);

  size_t lds_bytes = (size_t)10 * n * sizeof(float);   // 80 KB for n=2048
  (void)hipFuncSetAttribute((const void*)gs_render,
                            hipFuncAttributeMaxDynamicSharedMemorySize,
                            (int)lds_bytes);
  int waves = IMG_H * (IMG_W / 16);    // 1024 waves, 8 waves / block
  gs_render<<<waves / 8, 256, lds_bytes, stream>>>(
      ws + 10*(size_t)n, (float*)d_out, n);
}